// MoELinearHead_10797547782494
// MI455X (gfx1250) — compile-verified
//
#include <hip/hip_runtime.h>
#include <hip/hip_bf16.h>

// MoE linear head for MI455X (gfx1250, wave32).
// Phase 1 (gate) is HBM-bound (1 GiB feature stream); phases 2-4 are tiny.
// The classifier einsum is algebraically collapsed (sum_k comb_w == 1) into
// a single fp32 GEMM (256 x 8192) @ (8192 x 1000) executed with
// V_WMMA_F32_16X16X4_F32 so precision matches the fp32 reference exactly.
// GEMM hot loop: one global_load_b128 per WMMA operand feeds TWO wmma ops
// (K-order permuted consistently across A and B lanes).

#define BB 256
#define NN 1024
#define DD 1024
#define EE 8
#define LL 1000
#define TOPK 3

typedef __attribute__((ext_vector_type(2))) float v2f;
typedef __attribute__((ext_vector_type(4))) float v4f;
typedef __attribute__((ext_vector_type(8))) float v8f;

// ---------------------------------------------------------------------------
// Kernel 1: gate scores. One wave per (b,n) row, gate_w staged in LDS.
// Writes gate_score in (B, E, N) layout for coalesced softmax reads.
// ---------------------------------------------------------------------------
__global__ void __launch_bounds__(256)
gate_kernel(const float* __restrict__ feat,      // (B, N, D)
            const float* __restrict__ gate_w,    // (D, E)
            const float* __restrict__ gate_b,    // (E)
            float* __restrict__ gate_score)      // (B, E, N)
{
    __shared__ float wlds[DD * EE];              // 32 KB
    const int tid = threadIdx.x;
    for (int i = tid; i < DD * EE; i += 256) wlds[i] = gate_w[i];
    __syncthreads();

    const int lane = tid & 31;
    const int wave = tid >> 5;
    const long long row = (long long)blockIdx.x * 8 + wave;   // row in [0, B*N)

    const float4* __restrict__ frow =
        (const float4*)(feat + row * (long long)DD);

    float acc[EE];
#pragma unroll
    for (int e = 0; e < EE; ++e) acc[e] = 0.0f;

#pragma unroll
    for (int i = 0; i < DD / 128; ++i) {          // 8 iterations, b128 loads
        const float4 f = frow[i * 32 + lane];
        const int d = (i * 32 + lane) * 4;
#pragma unroll
        for (int e = 0; e < EE; ++e) {
            acc[e] += f.x * wlds[(d + 0) * EE + e]
                    + f.y * wlds[(d + 1) * EE + e]
                    + f.z * wlds[(d + 2) * EE + e]
                    + f.w * wlds[(d + 3) * EE + e];
        }
    }

    // wave32 butterfly reduction
#pragma unroll
    for (int e = 0; e < EE; ++e) {
        float v = acc[e];
        v += __shfl_xor(v, 16);
        v += __shfl_xor(v, 8);
        v += __shfl_xor(v, 4);
        v += __shfl_xor(v, 2);
        v += __shfl_xor(v, 1);
        acc[e] = v;
    }

    if (lane < EE) {
        const int b = (int)(row / NN);
        const int n = (int)(row % NN);
        gate_score[((long long)b * EE + lane) * NN + n] = acc[lane] + gate_b[lane];
    }
}

// ---------------------------------------------------------------------------
// Kernel 2: per-(b,e) softmax statistics over N + top-3 (+ comb_w softmax).
// top-k of softmax(prob) == top-k of raw scores (softmax is monotonic);
// only the 3 winning probabilities are materialized.
// ---------------------------------------------------------------------------
__global__ void __launch_bounds__(256)
topk_kernel(const float* __restrict__ gate_score,  // (B, E, N)
            int* __restrict__ top_i,               // (B, E, 3)
            float* __restrict__ comb_w)            // (B, E, 3)
{
    __shared__ float sv[NN];
    __shared__ float rv[256];
    __shared__ int   ri[256];
    __shared__ int   chosen[TOPK];
    __shared__ float chosenv[TOPK];

    const int tid = threadIdx.x;
    const long long base = (long long)blockIdx.x * NN;

    float lmax = -3.402823466e38f;
    for (int i = tid; i < NN; i += 256) {
        const float v = gate_score[base + i];
        sv[i] = v;
        lmax = fmaxf(lmax, v);
    }
    rv[tid] = lmax;
    __syncthreads();
    for (int off = 128; off; off >>= 1) {
        if (tid < off) rv[tid] = fmaxf(rv[tid], rv[tid + off]);
        __syncthreads();
    }
    const float M = rv[0];
    __syncthreads();

    float lsum = 0.0f;
    for (int i = tid; i < NN; i += 256) lsum += __expf(sv[i] - M);
    rv[tid] = lsum;
    __syncthreads();
    for (int off = 128; off; off >>= 1) {
        if (tid < off) rv[tid] += rv[tid + off];
        __syncthreads();
    }
    const float S = rv[0];
    __syncthreads();

    // 3 argmax passes (tie-break: lower index, matching lax.top_k)
    for (int t = 0; t < TOPK; ++t) {
        float bv = -3.402823466e38f;
        int   bi = 0x7fffffff;
        for (int i = tid; i < NN; i += 256) {
            bool skip = false;
            for (int u = 0; u < t; ++u)
                if (chosen[u] == i) skip = true;
            const float v = sv[i];
            if (!skip && (v > bv || (v == bv && i < bi))) { bv = v; bi = i; }
        }
        rv[tid] = bv;
        ri[tid] = bi;
        __syncthreads();
        for (int off = 128; off; off >>= 1) {
            if (tid < off) {
                const float ov = rv[tid + off];
                const int   oi = ri[tid + off];
                if (ov > rv[tid] || (ov == rv[tid] && oi < ri[tid])) {
                    rv[tid] = ov;
                    ri[tid] = oi;
                }
            }
            __syncthreads();
        }
        if (tid == 0) { chosen[t] = ri[0]; chosenv[t] = rv[0]; }
        __syncthreads();
    }

    if (tid == 0) {
        float p[TOPK];
        float pm = -3.402823466e38f;
#pragma unroll
        for (int t = 0; t < TOPK; ++t) {
            p[t] = __expf(chosenv[t] - M) / S;   // gate probability
            pm = fmaxf(pm, p[t]);
        }
        float ps = 0.0f;
#pragma unroll
        for (int t = 0; t < TOPK; ++t) { p[t] = __expf(p[t] - pm); ps += p[t]; }
#pragma unroll
        for (int t = 0; t < TOPK; ++t) {
            top_i[blockIdx.x * TOPK + t]  = chosen[t];
            comb_w[blockIdx.x * TOPK + t] = p[t] / ps;
        }
    }
}

// ---------------------------------------------------------------------------
// Kernel 3: Z[b,e,:] = sum_k comb_w[b,e,k] * features[b, top_i[b,e,k], :]
// One block per (b,e); 256 threads x float4 covers D=1024.
// ---------------------------------------------------------------------------
__global__ void __launch_bounds__(256)
combine_kernel(const float* __restrict__ feat,    // (B, N, D)
               const int* __restrict__ top_i,     // (B, E, 3)
               const float* __restrict__ comb_w,  // (B, E, 3)
               float* __restrict__ Z)             // (B, E, D) == (B, E*D)
{
    const int be = blockIdx.x;
    const int b  = be >> 3;
    const int tid = threadIdx.x;

    const int   i0 = top_i[be * 3 + 0];
    const int   i1 = top_i[be * 3 + 1];
    const int   i2 = top_i[be * 3 + 2];
    const float w0 = comb_w[be * 3 + 0];
    const float w1 = comb_w[be * 3 + 1];
    const float w2 = comb_w[be * 3 + 2];

    const float4* f0 = (const float4*)(feat + ((long long)b * NN + i0) * DD);
    const float4* f1 = (const float4*)(feat + ((long long)b * NN + i1) * DD);
    const float4* f2 = (const float4*)(feat + ((long long)b * NN + i2) * DD);

    const float4 a = f0[tid];
    const float4 c = f1[tid];
    const float4 d = f2[tid];
    float4 z;
    z.x = w0 * a.x + w1 * c.x + w2 * d.x;
    z.y = w0 * a.y + w1 * c.y + w2 * d.y;
    z.z = w0 * a.z + w1 * c.z + w2 * d.z;
    z.w = w0 * a.w + w1 * c.w + w2 * d.w;
    ((float4*)(Z + (long long)be * DD))[tid] = z;
}

// ---------------------------------------------------------------------------
// Kernel 4: out[b,l] = (1/E) * ( Z(b,:) . W(l,:) + sum_e cls_b[e,l] )
// where Z is (256 x 8192) and W(l, e*D+d) = cls_w[e,l,d].
// V_WMMA_F32_16X16X4_F32, one wave per 16x16 output tile, K kept in-wave
// (deterministic fp32 accumulation; no atomics).
//
// fp32 WMMA lane layout (ISA 7.12.2):
//   A 16x4 : lanes 0-15 -> M=lane,  V0=slotK0, V1=slotK1 ; lanes 16-31 -> slotK2/3
//   B 4x16 : lanes 0-15 -> N=lane,  V0=slotK0, V1=slotK1 ; lanes 16-31 -> slotK2/3
//   C 16x16: VGPR r, lanes 0-15 -> (M=r, N=lane) ; lanes 16-31 -> (M=r+8)
//
// K-permutation trick: per 8-wide K chunk, lanes 0-15 b128-load k=[8s..8s+3],
// lanes 16-31 b128-load k=[8s+4..8s+7]. wmma#1 consumes (x,y), wmma#2 (z,w).
// Each memory-k is consumed exactly once, with identical slot mapping on A
// and B, so the contraction is exact (fp32 sum order is free to permute up
// to rounding, same as any tiling choice).
// ---------------------------------------------------------------------------
__global__ void __launch_bounds__(256)
head_wmma_kernel(const float* __restrict__ Z,      // (B, E*D)
                 const float* __restrict__ cls_w,  // (E, L, D)
                 const float* __restrict__ cls_b,  // (E, L)
                 float* __restrict__ out)          // (B, L)
{
    const int tid  = threadIdx.x;
    const int lane = tid & 31;
    const int wave = tid >> 5;
    const int wg   = blockIdx.x * 8 + wave;    // 1008 tiles = 16 m x 63 n
    const int mt   = wg & 15;
    const int nt   = wg >> 4;
    const int m0   = mt * 16;
    const int n0   = nt * 16;

    const int half = lane >> 4;                // 0: k-slots {0,1}, 1: {2,3}
    const int l16  = lane & 15;

    const float* __restrict__ Arow = Z + (long long)(m0 + l16) * (EE * DD);
    int lcol = n0 + l16;
    if (lcol > LL - 1) lcol = LL - 1;          // clamp B-addr for the ragged tile

    v8f c = {0.f, 0.f, 0.f, 0.f, 0.f, 0.f, 0.f, 0.f};

    for (int e = 0; e < EE; ++e) {
        const float* __restrict__ Abase = Arow + e * DD + 4 * half;
        const float* __restrict__ Bbase =
            cls_w + (long long)e * LL * DD + (long long)lcol * DD + 4 * half;
#pragma unroll 4
        for (int dk = 0; dk < DD; dk += 8) {
            const v4f a4 = *(const v4f*)(Abase + dk);   // global_load_b128
            const v4f b4 = *(const v4f*)(Bbase + dk);   // global_load_b128
            const v2f a01 = __builtin_shufflevector(a4, a4, 0, 1);
            const v2f b01 = __builtin_shufflevector(b4, b4, 0, 1);
            c = __builtin_amdgcn_wmma_f32_16x16x4_f32(
                    false, a01, false, b01, (short)0, c, false, false);
            const v2f a23 = __builtin_shufflevector(a4, a4, 2, 3);
            const v2f b23 = __builtin_shufflevector(b4, b4, 2, 3);
            c = __builtin_amdgcn_wmma_f32_16x16x4_f32(
                    false, a23, false, b23, (short)0, c, false, false);
        }
    }

    // bias column: sum_e cls_b[e, l]
    float biasn = 0.0f;
#pragma unroll
    for (int e = 0; e < EE; ++e) biasn += cls_b[e * LL + lcol];

    const float scale = 1.0f / (float)EE;
    const int outn = n0 + l16;
    if (outn < LL) {
#pragma unroll
        for (int r = 0; r < 8; ++r) {
            const int m = m0 + r + half * 8;
            out[(long long)m * LL + outn] = (c[r] + biasn) * scale;
        }
    }
}

// ---------------------------------------------------------------------------
extern "C" void kernel_launch(void* const* d_in, const int* in_sizes, int n_in,
                              void* d_out, int out_size, void* d_ws, size_t ws_size,
                              hipStream_t stream) {
    const float* feat   = (const float*)d_in[0];   // (B, N, D)
    const float* gate_w = (const float*)d_in[1];   // (D, E)
    const float* gate_b = (const float*)d_in[2];   // (E)
    const float* cls_w  = (const float*)d_in[3];   // (E, L, D)
    const float* cls_b  = (const float*)d_in[4];   // (E, L)
    float* out = (float*)d_out;                    // (B, L)

    // workspace layout (~16.8 MB)
    char* ws = (char*)d_ws;
    const size_t score_bytes = (size_t)BB * EE * NN * sizeof(float);   // 8 MB
    const size_t z_bytes     = (size_t)BB * EE * DD * sizeof(float);   // 8 MB
    float* gate_score = (float*)ws;
    float* Zbuf       = (float*)(ws + score_bytes);
    int*   top_i      = (int*)  (ws + score_bytes + z_bytes);
    float* comb_w     = (float*)(ws + score_bytes + z_bytes
                                    + (size_t)BB * EE * TOPK * sizeof(int));

    // 1) gate scores: 32768 blocks x 8 waves = 262144 rows
    gate_kernel<<<(BB * NN) / 8, 256, 0, stream>>>(feat, gate_w, gate_b, gate_score);

    // 2) softmax stats + top-3 + comb_w: one block per (b,e)
    topk_kernel<<<BB * EE, 256, 0, stream>>>(gate_score, top_i, comb_w);

    // 3) weighted gather -> Z
    combine_kernel<<<BB * EE, 256, 0, stream>>>(feat, top_i, comb_w, Zbuf);

    // 4) WMMA GEMM: 16 m-tiles x 63 n-tiles = 1008 waves = 126 blocks
    head_wmma_kernel<<<((BB / 16) * ((LL + 15) / 16)) / 8, 256, 0, stream>>>(
        Zbuf, cls_w, cls_b, out);
}